// DAGNNConv_936302871055
// MI455X (gfx1250) — compile-verified
//
#include <hip/hip_runtime.h>
#include <hip/hip_bf16.h>

#define NNODES 10000
#define NEDGES 640000
#define CH     128
#define DEGPAD 10240          // deg region padded for 16B alignment of hop buffers
#define NTILES (NNODES / 16)  // 625 exact

typedef __attribute__((ext_vector_type(2))) float v2f;
typedef __attribute__((ext_vector_type(8))) float v8f;

// ---------------- workspace init ----------------
__global__ void dagnn_zero_u32(unsigned* __restrict__ p, int n) {
    int t = blockIdx.x * blockDim.x + threadIdx.x;
    if (t < n) p[t] = 0u;
}

// ---------------- in-degree count ----------------
__global__ void dagnn_count_deg(const int* __restrict__ dst, unsigned* __restrict__ deg, int ne) {
    int t = blockIdx.x * blockDim.x + threadIdx.x;
    if (t < ne) atomicAdd(&deg[dst[t]], 1u);
}

// ---------------- one propagation hop: hnext[dst] += hprev[src] ----------------
// 32 lanes per edge, one float4 (4 channels) per lane. Whole wave shares one edge
// -> src/dst loads are wave-uniform; gather is a coalesced 512B row (L2-resident).
__global__ void dagnn_scatter(const float* __restrict__ hprev, float* __restrict__ hnext,
                              const int* __restrict__ src, const int* __restrict__ dst) {
    int t = blockIdx.x * blockDim.x + threadIdx.x;
    int e = t >> 5;
    int q = t & 31;
    if (e >= NEDGES) return;
    int s = src[e];
    int d = dst[e];
    const float4 v = *reinterpret_cast<const float4*>(hprev + (size_t)s * CH + q * 4);
    float* p = hnext + (size_t)d * CH + q * 4;
    atomicAdd(p + 0, v.x);
    atomicAdd(p + 1, v.y);
    atomicAdd(p + 2, v.z);
    atomicAdd(p + 3, v.w);
}

// ---------------- mean normalization: h[n,:] *= (deg>0 ? 1/deg : 0) ----------------
__global__ void dagnn_normalize(float* __restrict__ h, const unsigned* __restrict__ deg) {
    int t = blockIdx.x * blockDim.x + threadIdx.x;
    int n = t >> 5;
    int q = t & 31;
    if (n >= NNODES) return;
    unsigned dg = deg[n];
    float inv = dg ? 1.0f / (float)dg : 0.0f;
    float4* p = reinterpret_cast<float4*>(h + (size_t)n * CH + q * 4);
    float4 v = *p;
    v.x *= inv; v.y *= inv; v.z *= inv; v.w *= inv;
    *p = v;
}

// ---------------- gated combine using V_WMMA_F32_16X16X4_F32 ----------------
// One wave32 per 16-node tile. scores[16 nodes][4 hops] computed by 4 chains of
// 32 WMMAs (K=4 f32 chunks over 128 channels); B = gate_w chunk replicated in
// every column so D[m,*] = dot(h_hop[m,:], gate_w). Then softmax over the 4 hop
// scores and a weighted sum of the LDS-staged hop features.
__global__ void dagnn_combine(const float* __restrict__ x,
                              const float* __restrict__ h1,
                              const float* __restrict__ h2,
                              const float* __restrict__ h3,
                              const float* __restrict__ gate_w,
                              const float* __restrict__ gate_b,
                              float* __restrict__ out) {
    __shared__ __align__(16) float Hl[4][16][CH];  // 32 KB
    __shared__ float gw[CH];
    __shared__ float wl[16][4];

    const int lane = threadIdx.x;            // 0..31
    const size_t base = (size_t)blockIdx.x * 16 * CH;
    const float* hp0 = x  + base;
    const float* hp1 = h1 + base;
    const float* hp2 = h2 + base;
    const float* hp3 = h3 + base;

    for (int i = lane; i < CH; i += 32) gw[i] = gate_w[i];

    const float4* s0 = reinterpret_cast<const float4*>(hp0);
    const float4* s1 = reinterpret_cast<const float4*>(hp1);
    const float4* s2 = reinterpret_cast<const float4*>(hp2);
    const float4* s3 = reinterpret_cast<const float4*>(hp3);
    float4* l0 = reinterpret_cast<float4*>(&Hl[0][0][0]);
    float4* l1 = reinterpret_cast<float4*>(&Hl[1][0][0]);
    float4* l2 = reinterpret_cast<float4*>(&Hl[2][0][0]);
    float4* l3 = reinterpret_cast<float4*>(&Hl[3][0][0]);
    for (int i = lane; i < 16 * CH / 4; i += 32) {
        l0[i] = s0[i];
        l1[i] = s1[i];
        l2[i] = s2[i];
        l3[i] = s3[i];
    }
    __syncthreads();

    const int m  = lane & 15;   // A-matrix row M for this lane
    const int hi = lane >> 4;   // half-wave selects K pair {0,1} vs {2,3}

    v8f acc[4] = {};            // one f32 16x16 accumulator per hop

#pragma unroll 4
    for (int i = 0; i < 32; ++i) {
        const int c0 = 4 * i + 2 * hi;
        v2f b;
        b.x = gw[c0];
        b.y = gw[c0 + 1];
#pragma unroll
        for (int k = 0; k < 4; ++k) {
            v2f a;
            a.x = Hl[k][m][c0];
            a.y = Hl[k][m][c0 + 1];
            acc[k] = __builtin_amdgcn_wmma_f32_16x16x4_f32(
                /*neg_a=*/false, a, /*neg_b=*/false, b,
                /*c_mod=*/(short)0, acc[k],
                /*reuse_a=*/false, /*reuse_b=*/false);
        }
    }

    // lane l holds scores for nodes r + 8*hi in acc[k][r]; lanes m==0 of each
    // half-wave publish the softmax weights for their 8 nodes.
    if (m == 0) {
        const float gb = gate_b[0];
#pragma unroll
        for (int r = 0; r < 8; ++r) {
            const int node = r + 8 * hi;
            float sc0 = acc[0][r] + gb;
            float sc1 = acc[1][r] + gb;
            float sc2 = acc[2][r] + gb;
            float sc3 = acc[3][r] + gb;
            float mx = fmaxf(fmaxf(sc0, sc1), fmaxf(sc2, sc3));
            float e0 = expf(sc0 - mx);
            float e1 = expf(sc1 - mx);
            float e2 = expf(sc2 - mx);
            float e3 = expf(sc3 - mx);
            float inv = 1.0f / (e0 + e1 + e2 + e3);
            wl[node][0] = e0 * inv;
            wl[node][1] = e1 * inv;
            wl[node][2] = e2 * inv;
            wl[node][3] = e3 * inv;
        }
    }
    __syncthreads();

    float4* outp = reinterpret_cast<float4*>(out + base);
    for (int i = lane; i < 16 * CH / 4; i += 32) {
        const int node = i >> 5;  // 32 float4s per 128-ch row
        float w0 = wl[node][0], w1 = wl[node][1], w2 = wl[node][2], w3 = wl[node][3];
        float4 a = l0[i], b = l1[i], c = l2[i], d = l3[i];
        float4 o;
        o.x = w0 * a.x + w1 * b.x + w2 * c.x + w3 * d.x;
        o.y = w0 * a.y + w1 * b.y + w2 * c.y + w3 * d.y;
        o.z = w0 * a.z + w1 * b.z + w2 * c.z + w3 * d.z;
        o.w = w0 * a.w + w1 * b.w + w2 * c.w + w3 * d.w;
        outp[i] = o;
    }
}

extern "C" void kernel_launch(void* const* d_in, const int* in_sizes, int n_in,
                              void* d_out, int out_size, void* d_ws, size_t ws_size,
                              hipStream_t stream) {
    const float* x      = (const float*)d_in[0];
    const int*   ei     = (const int*)d_in[1];   // [2, NEDGES]
    const float* gate_w = (const float*)d_in[2];
    const float* gate_b = (const float*)d_in[3];
    const int* src = ei;
    const int* dst = ei + NEDGES;

    unsigned* deg = (unsigned*)d_ws;
    float* h1 = (float*)d_ws + DEGPAD;
    float* h2 = h1 + (size_t)NNODES * CH;
    float* h3 = h2 + (size_t)NNODES * CH;
    float* outf = (float*)d_out;

    // zero deg + all three hop buffers (ws is poisoned; must re-init every call)
    const int zcount = DEGPAD + 3 * NNODES * CH;
    dagnn_zero_u32<<<(zcount + 255) / 256, 256, 0, stream>>>((unsigned*)d_ws, zcount);

    dagnn_count_deg<<<(NEDGES + 255) / 256, 256, 0, stream>>>(dst, deg, NEDGES);

    const int scatterThreads = NEDGES * 32;
    const int normThreads = NNODES * 32;

    dagnn_scatter<<<(scatterThreads + 255) / 256, 256, 0, stream>>>(x, h1, src, dst);
    dagnn_normalize<<<(normThreads + 255) / 256, 256, 0, stream>>>(h1, deg);

    dagnn_scatter<<<(scatterThreads + 255) / 256, 256, 0, stream>>>(h1, h2, src, dst);
    dagnn_normalize<<<(normThreads + 255) / 256, 256, 0, stream>>>(h2, deg);

    dagnn_scatter<<<(scatterThreads + 255) / 256, 256, 0, stream>>>(h2, h3, src, dst);
    dagnn_normalize<<<(normThreads + 255) / 256, 256, 0, stream>>>(h3, deg);

    dagnn_combine<<<NTILES, 32, 0, stream>>>(x, h1, h2, h3, gate_w, gate_b, outf);
}